// LMModel_89232240542569
// MI455X (gfx1250) — compile-verified
//
#include <hip/hip_runtime.h>

typedef __attribute__((ext_vector_type(2))) float v2f;
typedef __attribute__((ext_vector_type(8))) float v8f;

#define KC      64   // K chunk staged in LDS per iteration
#define LDA_PAD 68   // 64 + 4 floats: A rows land on distinct LDS banks

// ---------------------------------------------------------------------------
// Fused WMMA f32 GEMM:  out = POST( ACT( pre + A1@B1 [+ A2@B2] + bias ) )
//   A: row-major [M][KD].   B: row-major [KD][LDB] (LDB compile-time so the
//   unrolled chunk's B loads become immediate-offset global_load_b32).
//   Block: 256 threads = 8 waves. Block tile = (RT*16) rows x 128 cols.
//   Wave w owns cols [bx*128 + w*16, +16) and RT register row-tiles, so each
//   B fragment is reused RT times. A tiles staged via LDS (ds_load_2addr_b64).
//
// ACT : 0 none, 1 sigmoid, 2 tanh
// POST: 0 out[gm*ldo+gn] = val
//       1 fused r/z (requires GATED): gate0 -> out(rh) = val * hmat ;
//                                     gate1 -> hbuf(z) = val
//       2 GRU blend: hbuf = (1-z)*hbuf + z*val ; optionally copied to out2
// GATED: column space spans gates of 1024 cols; gate selects B1/B2 (+g*1024^2),
//        bias (+g*1024) and pre (+g*preGateStride).
// ---------------------------------------------------------------------------
template<int ACT, int POST, bool HAS2, int RT, bool GATED, int LDB, int KD>
__global__ __launch_bounds__(256) void gemm_wmma(
    const float* __restrict__ A1, const float* __restrict__ B1,
    const float* __restrict__ A2, const float* __restrict__ B2,
    const float* __restrict__ pre, int ldpre, size_t preGateStride,
    const float* __restrict__ bias,
    const float* __restrict__ hmat,
    const float* __restrict__ zvec,
    float*       hbuf,
    float* __restrict__ out, int ldo,
    float*       out2,
    int M, int N)
{
  __shared__ __align__(16) float sA1[RT * 16 * LDA_PAD];
  __shared__ __align__(16) float sA2[HAS2 ? RT * 16 * LDA_PAD : 4];

  const int tid  = threadIdx.x;
  const int wave = tid >> 5;
  const int lane = tid & 31;
  const int rr   = lane & 15;          // M-row (A) / N-col (B) within a tile
  const int koff = (lane >> 4) << 1;   // lanes 0-15 -> K{0,1}; 16-31 -> K{2,3}
  const int row0 = blockIdx.y * (RT * 16);
  const int col0 = blockIdx.x * 128 + wave * 16;

  // Gate-uniform operand selection (uniform per wave -> scalar branches).
  int gate = 0, nl0 = col0;
  if constexpr (GATED) { gate = col0 >> 10; nl0 = col0 & 1023; }
  const float* B1p  = GATED ? B1 + (size_t)gate * 1024 * 1024 : B1;
  const float* B2p  = (HAS2 && GATED) ? B2 + (size_t)gate * 1024 * 1024 : B2;
  const float* bsp  = (bias && GATED) ? bias + (size_t)gate * 1024 : bias;
  const float* prep = (pre  && GATED) ? pre  + (size_t)gate * preGateStride : pre;

  v8f acc[RT];
#pragma unroll
  for (int rt = 0; rt < RT; ++rt) acc[rt] = (v8f){};

  for (int kc = 0; kc < KD; kc += KC) {
    __syncthreads();
    // Stage A tile(s): (RT*16) rows x 64 cols, float4 cooperative loads.
    for (int i = tid; i < RT * 16 * (KC / 4); i += 256) {
      const int r = i >> 4, c = i & 15;
      float4 v = *(const float4*)(A1 + (size_t)(row0 + r) * KD + kc + c * 4);
      *(float4*)&sA1[r * LDA_PAD + c * 4] = v;
      if constexpr (HAS2) {
        float4 v2 = *(const float4*)(A2 + (size_t)(row0 + r) * KD + kc + c * 4);
        *(float4*)&sA2[r * LDA_PAD + c * 4] = v2;
      }
    }
    __syncthreads();

    // Per-lane B base pointers for this chunk; all loads inside the unrolled
    // loop below are at compile-time (k*LDB) offsets -> immediate IOFFSETs.
    const float* bp1 = B1p + (size_t)(kc + koff) * LDB + nl0 + rr;
    const float* bp2 = HAS2 ? B2p + (size_t)(kc + koff) * LDB + nl0 + rr : nullptr;

    if (kc + KC < KD) {
      __builtin_prefetch(bp1 + (size_t)KC * LDB, 0, 3);
      if constexpr (HAS2) __builtin_prefetch(bp2 + (size_t)KC * LDB, 0, 3);
    }

#pragma unroll
    for (int k = 0; k < KC; k += 4) {
      v2f b1;
      b1.x = bp1[(size_t)k * LDB];
      b1.y = bp1[(size_t)k * LDB + LDB];
      v2f b2;
      if constexpr (HAS2) {
        b2.x = bp2[(size_t)k * LDB];
        b2.y = bp2[(size_t)k * LDB + LDB];
      }
#pragma unroll
      for (int rt = 0; rt < RT; ++rt) {
        v2f a1;
        const int ab = (rt * 16 + rr) * LDA_PAD + k + koff;
        a1.x = sA1[ab];
        a1.y = sA1[ab + 1];
        acc[rt] = __builtin_amdgcn_wmma_f32_16x16x4_f32(false, a1, false, b1,
                                                        (short)0, acc[rt], false, false);
        if constexpr (HAS2) {
          v2f a2;
          a2.x = sA2[ab];
          a2.y = sA2[ab + 1];
          acc[rt] = __builtin_amdgcn_wmma_f32_16x16x4_f32(false, a2, false, b2,
                                                          (short)0, acc[rt], false, false);
        }
      }
    }
  }

  // Epilogue. C/D layout: VGPR i -> M = i (+8 for lanes 16-31), N = lane%16.
  const bool hi = lane >= 16;
  const int  gn = col0 + rr;      // global col in launch space
  const int  nl = nl0 + rr;       // col within the 1024-wide gate
#pragma unroll
  for (int rt = 0; rt < RT; ++rt) {
#pragma unroll
    for (int i = 0; i < 8; ++i) {
      const int gm = row0 + rt * 16 + (hi ? 8 : 0) + i;
      float val = acc[rt][i];
      if (prep) val += prep[(size_t)gm * ldpre + nl];
      if (bsp)  val += bsp[nl];
      if constexpr (ACT == 1) val = 1.0f / (1.0f + expf(-val));
      else if constexpr (ACT == 2) val = tanhf(val);

      if constexpr (POST == 0) {
        out[(size_t)gm * ldo + gn] = val;
      } else if constexpr (POST == 1) {
        if (gate == 0) out[(size_t)gm * 1024 + nl] = val * hmat[(size_t)gm * 1024 + nl];
        else           hbuf[(size_t)gm * 1024 + nl] = val;   // z
      } else {
        const float z = zvec[(size_t)gm * 1024 + nl];
        const float h = hbuf[(size_t)gm * 1024 + nl];
        const float hn = (1.0f - z) * h + z * val;
        hbuf[(size_t)gm * 1024 + nl] = hn;
        if (out2) out2[(size_t)gm * 1024 + nl] = hn;
      }
    }
  }
}

// ---------------------------------------------------------------------------
__global__ void embed_gather(const int* __restrict__ tokens,
                             const float* __restrict__ E,
                             float* __restrict__ emb)
{
  const int sb  = blockIdx.x;
  const int tok = tokens[sb];
  const float4* src = (const float4*)(E   + (size_t)tok * 1024);
  float4*       dst = (float4*)      (emb + (size_t)sb  * 1024);
  for (int i = threadIdx.x; i < 256; i += blockDim.x) dst[i] = src[i];
}

__global__ void copy2_f32(const float* __restrict__ a, const float* __restrict__ b,
                          float* __restrict__ oa, float* __restrict__ ob, int n)
{
  const int i = blockIdx.x * blockDim.x + threadIdx.x;
  if (i < n) { oa[i] = a[i]; ob[i] = b[i]; }
}

__global__ void add_f32(const float* __restrict__ a, const float* __restrict__ b,
                        float* __restrict__ o, int n)
{
  const int i = blockIdx.x * blockDim.x + threadIdx.x;
  if (i < n) o[i] = a[i] + b[i];
}

// ---------------------------------------------------------------------------
extern "C" void kernel_launch(void* const* d_in, const int* in_sizes, int n_in,
                              void* d_out, int out_size, void* d_ws, size_t ws_size,
                              hipStream_t stream)
{
  (void)in_sizes; (void)n_in; (void)out_size; (void)ws_size;
  constexpr int SEQ = 128, BATCH = 32, NVOC = 32000, D = 1024;
  constexpr int SB = SEQ * BATCH;                 // 4096
  constexpr size_t DD = (size_t)D * D;

  const int*   tokens = (const int*)  d_in[0];
  const float* E      = (const float*)d_in[1];
  const float* Wx1    = (const float*)d_in[2];
  const float* bx1    = (const float*)d_in[3];
  const float* Wh1    = (const float*)d_in[4];
  const float* bh1    = (const float*)d_in[5];
  const float* Wx2    = (const float*)d_in[6];
  const float* bx2    = (const float*)d_in[7];
  const float* Wh2    = (const float*)d_in[8];
  const float* bh2    = (const float*)d_in[9];
  const float* Wd     = (const float*)d_in[10];
  const float* bd     = (const float*)d_in[11];
  const float* h1i    = (const float*)d_in[12];
  const float* h2i    = (const float*)d_in[13];
  float* out = (float*)d_out;

  // Workspace layout (floats). Total ~81 MB.
  float* ws = (float*)d_ws;
  size_t off = 0;
  float* emb   = ws + off; off += (size_t)SB * D;        // 16 MB
  float* X1    = ws + off; off += (size_t)3 * SB * D;    // 48 MB (gate-major)
  float* h2seq = ws + off; off += (size_t)SB * D;        // 16 MB
  float* h1    = ws + off; off += (size_t)BATCH * D;
  float* h2    = ws + off; off += (size_t)BATCH * D;
  float* rh    = ws + off; off += (size_t)BATCH * D;
  float* zb    = ws + off; off += (size_t)BATCH * D;
  float* bsum2 = ws + off; off += (size_t)3 * D;

  const dim3 blk(256);
  const dim3 gRZ(2 * D / 128, 1);          // (16,1)  fused r+z, M=32 (RT=2)
  const dim3 gC(D / 128, 1);               // (8,1)   c gate,    M=32 (RT=2)
  const dim3 gPre(D / 128, SB / 64);       // (8,64)  X1 precompute (RT=4)
  const dim3 gDec(NVOC / 128, SB / 64);    // (250,64) decoder (RT=4)

  // Init: copy hidden state into ws (inputs immutable); bsum2 = bx2 + bh2.
  copy2_f32<<<dim3((BATCH * D + 255) / 256), dim3(256), 0, stream>>>(h1i, h2i, h1, h2, BATCH * D);
  add_f32<<<dim3((3 * D + 255) / 256), dim3(256), 0, stream>>>(bx2, bh2, bsum2, 3 * D);

  // Embedding lookup.
  embed_gather<<<dim3(SB), dim3(256), 0, stream>>>(tokens, E, emb);

  // Hoisted layer-1 input projections: X1[g] = emb @ Wx1[g] + bx1[g]  (M=4096).
  for (int g = 0; g < 3; ++g) {
    gemm_wmma<0, 0, false, 4, false, 1024, 1024><<<gPre, blk, 0, stream>>>(
        emb, Wx1 + (size_t)g * DD, nullptr, nullptr,
        nullptr, 0, 0, bx1 + (size_t)g * D,
        nullptr, nullptr, nullptr,
        X1 + (size_t)g * SB * D, D, nullptr, SB, D);
  }

  // Sequential scan: 4 fused GEMMs per timestep, stream-ordered.
  for (int t = 0; t < SEQ; ++t) {
    const float* pt = X1 + (size_t)t * BATCH * D;   // gate stride = SB*D

    // Layer 1: fused r,z  (gated N=2048): rh = sigm(.)*h1 ; zb = sigm(.)
    gemm_wmma<1, 1, false, 2, true, 1024, 1024><<<gRZ, blk, 0, stream>>>(
        h1, Wh1, nullptr, nullptr,
        pt, D, (size_t)SB * D, bh1,
        h1, nullptr, zb, rh, D, nullptr, BATCH, 2 * D);
    // Layer 1: c + blend -> h1 (in place)
    gemm_wmma<2, 2, false, 2, false, 1024, 1024><<<gC, blk, 0, stream>>>(
        rh, Wh1 + 2 * DD, nullptr, nullptr,
        pt + (size_t)2 * SB * D, D, 0, bh1 + 2 * D,
        nullptr, zb, h1, nullptr, D, nullptr, BATCH, D);

    // Layer 2: fused r,z with fused x-projection (+ h1 @ Wx2[g])
    gemm_wmma<1, 1, true, 2, true, 1024, 1024><<<gRZ, blk, 0, stream>>>(
        h2, Wh2, h1, Wx2,
        nullptr, 0, 0, bsum2,
        h2, nullptr, zb, rh, D, nullptr, BATCH, 2 * D);
    // Layer 2: c + blend -> h2, copy to h2seq[t]
    gemm_wmma<2, 2, true, 2, false, 1024, 1024><<<gC, blk, 0, stream>>>(
        rh, Wh2 + 2 * DD, h1, Wx2 + 2 * DD,
        nullptr, 0, 0, bsum2 + 2 * D,
        nullptr, zb, h2, nullptr, D, h2seq + (size_t)t * BATCH * D, BATCH, D);
  }

  // Decoder: out = h2seq @ Wd + bd   (M=4096, N=32000, K=1024).
  gemm_wmma<0, 0, false, 4, false, 32000, 1024><<<gDec, blk, 0, stream>>>(
      h2seq, Wd, nullptr, nullptr,
      nullptr, 0, 0, bd,
      nullptr, nullptr, nullptr,
      out, NVOC, nullptr, SB, NVOC);
}